// Voxels_61684320305334
// MI455X (gfx1250) — compile-verified
//
#include <hip/hip_runtime.h>

// ---------------------------------------------------------------------------
// Voxels: Gaussian occupancy splat, B x N atoms -> (B, C, bx, by, bz) volume.
// MI455X reasoning: 31.9M voxel updates, each = ~20 f32 VALU + 1 v_exp_f32 +
// one 4B global_atomic_add_f32. Volume (~45MB) fits in the 192MB L2, so the
// kernel is L2-atomic bound; HBM is irrelevant (final 45MB flush = ~2us at
// 23.3 TB/s; VALU/exp work ~2us). No GEMM structure -> WMMA does not apply;
// the CDNA5-specific paths exercised are wave32 codegen, no-return f32 global
// atomics, v_exp_f32 transcendental co-issue (dual-issue VALU), and
// WGP-scope global_prefetch_b8.
// ---------------------------------------------------------------------------

#define BATCH   4
#define CADIM   5
#define LATO    11                 // 2*CAD+1
#define KVOX    (LATO*LATO*LATO)   // 1331
#define RESOL   1.0f
#define SIGMA_C 0.93f

__device__ __forceinline__ float block_min256(float v, float* sm) {
    const int t = threadIdx.x;
    sm[t] = v; __syncthreads();
    #pragma unroll
    for (int s = 128; s > 0; s >>= 1) {
        if (t < s) sm[t] = fminf(sm[t], sm[t + s]);
        __syncthreads();
    }
    float r = sm[0]; __syncthreads();
    return r;
}

__device__ __forceinline__ float block_max256(float v, float* sm) {
    const int t = threadIdx.x;
    sm[t] = v; __syncthreads();
    #pragma unroll
    for (int s = 128; s > 0; s >>= 1) {
        if (t < s) sm[t] = fmaxf(sm[t], sm[t + s]);
        __syncthreads();
    }
    float r = sm[0]; __syncthreads();
    return r;
}

// One block, 256 threads (8 wave32s). Computes per-batch trunc'd mincoords
// (the translation bases) and the integer box dimensions, writing them to ws.
__global__ __launch_bounds__(256)
void minmax_box_kernel(const float* __restrict__ coords, int N,
                       float* __restrict__ ws_minc /* [BATCH*3] */,
                       int* __restrict__ ws_box    /* [3]      */) {
    __shared__ float sm[256];
    const int t = threadIdx.x;

    float bmin[BATCH][3];
    float gmax0 = -3.4e38f, gmax1 = -3.4e38f, gmax2 = -3.4e38f;

    #pragma unroll
    for (int b = 0; b < BATCH; ++b) {
        const float* base = coords + (size_t)b * 3u * (size_t)N;
        float m0 = 3.4e38f, m1 = 3.4e38f, m2 = 3.4e38f;
        for (int n = t; n < N; n += 256) {
            // gfx1250 global_prefetch_b8, WGP scope (fills all cache levels;
            // OOB speculative prefetch is silently dropped per ISA 10.5)
            __builtin_prefetch(base + 3 * (n + 1024), 0, 3);
            float x = base[3 * n + 0];
            float y = base[3 * n + 1];
            float z = base[3 * n + 2];
            m0 = fminf(m0, x); m1 = fminf(m1, y); m2 = fminf(m2, z);
            gmax0 = fmaxf(gmax0, x); gmax1 = fmaxf(gmax1, y); gmax2 = fmaxf(gmax2, z);
        }
        bmin[b][0] = m0; bmin[b][1] = m1; bmin[b][2] = m2;
    }

    float rmin[BATCH][3];
    #pragma unroll
    for (int b = 0; b < BATCH; ++b) {
        #pragma unroll
        for (int d = 0; d < 3; ++d)
            rmin[b][d] = block_min256(bmin[b][d], sm);
    }
    float rmax[3];
    rmax[0] = block_max256(gmax0, sm);
    rmax[1] = block_max256(gmax1, sm);
    rmax[2] = block_max256(gmax2, sm);

    if (t == 0) {
        float minm[3] = {3.4e38f, 3.4e38f, 3.4e38f};
        #pragma unroll
        for (int b = 0; b < BATCH; ++b) {
            #pragma unroll
            for (int d = 0; d < 3; ++d) {
                float mc = truncf(rmin[b][d] * (1.0f / RESOL)); // trunc(min/RES)
                ws_minc[b * 3 + d] = mc;                        // translation base
                minm[d] = fminf(minm[d], mc);
            }
        }
        #pragma unroll
        for (int d = 0; d < 3; ++d) {
            // int(ceil(maxc/RES) - minm + (2*CAD+1)); whole numbers, exact in f32
            ws_box[d] = (int)(ceilf(rmax[d] * (1.0f / RESOL)) - minm[d]
                              + (float)(2 * CADIM + 1));
        }
    }
}

// 16B-vectorized clear: out_size is divisible by B*C=32, so the float4 body
// covers everything; scalar tail kept defensively. One global_store_b128 per
// lane -> 512B per wave32, saturates L2 write bandwidth with minimal issue.
__global__ __launch_bounds__(256)
void zero_kernel(float4* __restrict__ p4, long long n4,
                 float* __restrict__ p, long long n) {
    long long i = (long long)blockIdx.x * 256 + threadIdx.x;
    if (i < n4) {
        p4[i] = make_float4(0.0f, 0.0f, 0.0f, 0.0f);
    }
    long long tail = 4 * n4 + i;
    if (tail < n) p[tail] = 0.0f;
}

// grid = (ceil(N*KVOX/256), BATCH). One thread per (atom, voxel-of-cube).
// Consecutive lanes hit consecutive z addresses (fastest dim) -> coalesced
// no-return global_atomic_add_f32 that stays resident in the 192MB L2.
__global__ __launch_bounds__(256)
void voxelize_kernel(const float* __restrict__ coords,
                     const float* __restrict__ radius,
                     const int*   __restrict__ channels,
                     const int*   __restrict__ nchan,
                     int N,
                     const float* __restrict__ ws_minc,
                     const int*   __restrict__ ws_box,
                     float* __restrict__ vol) {
    const int b  = blockIdx.y;
    const int gx = blockIdx.x * 256 + threadIdx.x;
    const int per_batch = N * KVOX;
    if (gx >= per_batch) return;

    const int a = gx / KVOX;            // magic-multiply (KVOX is constant)
    const int k = gx - a * KVOX;
    const int atom = b * N + a;

    const int C  = *nchan;
    const int bx = ws_box[0];
    const int by = ws_box[1];
    const int bz = ws_box[2];

    const float tx = ws_minc[b * 3 + 0] - (float)CADIM;  // translation
    const float ty = ws_minc[b * 3 + 1] - (float)CADIM;
    const float tz = ws_minc[b * 3 + 2] - (float)CADIM;

    const float* p = coords + (size_t)atom * 3u;
    const float csx = p[0] * (1.0f / RESOL) - tx;        // scaled coords
    const float csy = p[1] * (1.0f / RESOL) - ty;
    const float csz = p[2] * (1.0f / RESOL) - tz;

    const float dscx = truncf(csx);                      // discrete coords
    const float dscy = truncf(csy);
    const float dscz = truncf(csz);

    const int kz = k % LATO;                             // constant divisors
    const int kt = k / LATO;
    const int ky = kt % LATO;
    const int kx = kt / LATO;
    const float ox = (float)(kx - CADIM);
    const float oy = (float)(ky - CADIM);
    const float oz = (float)(kz - CADIM);

    // voxel center = disc + offset + 1 + 0.5*RES
    const float half = 1.0f + 0.5f * RESOL;
    const float dx = csx - (dscx + ox + half);
    const float dy = csy - (dscy + oy + half);
    const float dz = csz - (dscz + oz + half);

    const float r    = radius[atom] * (1.0f / RESOL);
    const float den  = (SIGMA_C * SIGMA_C) * r * r;
    const float expo = -(dx * dx + dy * dy + dz * dz) / den;
    const float occ  = __expf(fminf(expo, 10.0f));       // v_exp_f32; expo <= 0

    const int ix = (int)dscx + kx - CADIM + 1;           // disc + cube + 1
    const int iy = (int)dscy + ky - CADIM + 1;
    const int iz = (int)dscz + kz - CADIM + 1;
    const int c  = channels[atom];

    const size_t off = ((((size_t)b * C + c) * bx + ix) * by + iy) * (size_t)bz + iz;
    atomicAdd(vol + off, occ);                           // no-return f32 atomic
}

extern "C" void kernel_launch(void* const* d_in, const int* in_sizes, int n_in,
                              void* d_out, int out_size, void* d_ws, size_t ws_size,
                              hipStream_t stream) {
    const float* coords   = (const float*)d_in[0];  // (B, N, 3) f32
    const float* radius   = (const float*)d_in[1];  // (B, N)    f32
    const int*   channels = (const int*)  d_in[2];  // (B, N)    i32
    const int*   nchan    = (const int*)  d_in[3];  // scalar    i32

    const int N = in_sizes[1] / BATCH;

    float* ws_minc = (float*)d_ws;            // BATCH*3 floats
    int*   ws_box  = (int*)d_ws + 32;         // 3 ints at byte offset 128

    // 1) data-dependent translation + box dims
    minmax_box_kernel<<<1, 256, 0, stream>>>(coords, N, ws_minc, ws_box);

    // 2) zero the accumulator (harness poisons d_out; must re-zero every call)
    const long long n_out = (long long)out_size;
    const long long n4    = n_out / 4;
    const long long zthreads = (n4 > 0 ? n4 : 1);
    zero_kernel<<<(unsigned)((zthreads + 255) / 256), 256, 0, stream>>>(
        (float4*)d_out, n4, (float*)d_out, n_out);

    // 3) scatter-add splat: one thread per (atom, cube voxel)
    const long long per_batch = (long long)N * KVOX;
    dim3 grid((unsigned)((per_batch + 255) / 256), BATCH);
    voxelize_kernel<<<grid, 256, 0, stream>>>(coords, radius, channels, nchan,
                                              N, ws_minc, ws_box, (float*)d_out);
}